// LQG_49478023250271
// MI455X (gfx1250) — compile-verified
//
#include <hip/hip_runtime.h>

typedef float v2f __attribute__((ext_vector_type(2)));
typedef float v8f __attribute__((ext_vector_type(8)));

#define SD   256   // fused state dim [x; xhat]
#define NDIM 128
#define MDIM 32
#define PDIM 64
#define TLEN 4096
#define BS   16    // steps per block
#define NB   256   // TLEN / BS

// ---------------------------------------------------------------------------
// Generic fp32 WMMA GEMM:  C = alpha * (op(A) @ B) [+ addScale * Cadd]
// One 16x16 output tile per wave, K-loop in steps of 4 using
// V_WMMA_F32_16X16X4_F32.  M,N multiples of 16.  K / transA / addend are
// compile-time so the hot loop has no branches and unrolls cleanly.
// ---------------------------------------------------------------------------
template <int KD, bool TA, bool ADDC>
__global__ void wgemm_t(const float* __restrict__ A, int lda,
                        const float* __restrict__ B, int ldb,
                        float* __restrict__ C, int ldc,
                        const float* __restrict__ Cadd, int ldadd, float addScale,
                        int Mdim, int Ndim, float alpha)
{
    const int lane = threadIdx.x & 31;
    const int wave = threadIdx.x >> 5;
    const int wavesPerBlk = blockDim.x >> 5;
    const int tile = blockIdx.x * wavesPerBlk + wave;
    const int tilesN = Ndim >> 4;
    const int tilesTot = (Mdim >> 4) * tilesN;
    if (tile >= tilesTot) return;          // uniform per wave: EXEC stays all-ones

    const int m0   = (tile / tilesN) << 4;
    const int n0   = (tile % tilesN) << 4;
    const int half = lane >> 4;            // 0: lanes 0-15, 1: lanes 16-31
    const int lm   = lane & 15;

    const float* __restrict__ Arow = A + (TA ? (m0 + lm) : (m0 + lm) * lda);
    const float* __restrict__ Bcol = B + (n0 + lm);

    v8f acc = {};
#pragma unroll 8
    for (int k0 = 0; k0 < KD; k0 += 4) {
        // A 16x4 fp32 layout: VGPR0 holds K=k0 (lanes 0-15) / K=k0+2 (lanes 16-31),
        // VGPR1 holds K=k0+1 / K=k0+3  (ISA 7.12.2)
        const int ka = k0 + (half << 1);
        v2f a, b;
        if constexpr (TA) {
            a[0] = Arow[(ka    ) * lda];
            a[1] = Arow[(ka + 1) * lda];
        } else {
            a = *reinterpret_cast<const v2f*>(Arow + ka);   // global_load_b64
        }
        b[0] = Bcol[(ka    ) * ldb];
        b[1] = Bcol[(ka + 1) * ldb];
        acc = __builtin_amdgcn_wmma_f32_16x16x4_f32(false, a, false, b,
                                                    (short)0, acc, false, false);
    }
    // C/D layout: VGPR r -> M = r (lanes 0-15) / r+8 (lanes 16-31), N = lane%16
#pragma unroll
    for (int r = 0; r < 8; ++r) {
        const int row = m0 + r + (half << 3);
        const int col = n0 + lm;
        float v = alpha * acc[r];
        if constexpr (ADDC) v += addScale * Cadd[row * ldadd + col];
        C[row * ldc + col] = v;
    }
}

// ---------------------------------------------------------------------------
// Assemble fused transition F (256x256), its transpose FT, and drive vector g.
//   F = [[ A        , -B K                      ],
//        [ L C A    , (A - LC) - (I + LC) B K   ]]
//   g = [ 1 ; LC @ 1 ]
// ---------------------------------------------------------------------------
__global__ void assembleF(const float* __restrict__ A,  const float* __restrict__ LC,
                          const float* __restrict__ BK, const float* __restrict__ LCA,
                          const float* __restrict__ IBK,
                          float* __restrict__ F, float* __restrict__ FT,
                          float* __restrict__ g)
{
    const int idx = blockIdx.x * blockDim.x + threadIdx.x;   // 0 .. 65535
    const int r = idx >> 8, c = idx & 255;
    float v;
    if (r < NDIM) {
        v = (c < NDIM) ? A[r * NDIM + c] : -BK[r * NDIM + (c - NDIM)];
    } else {
        const int i = r - NDIM;
        if (c < NDIM) {
            v = LCA[i * NDIM + c];
        } else {
            const int j = c - NDIM;
            v = A[i * NDIM + j] - LC[i * NDIM + j] - IBK[i * NDIM + j];
        }
    }
    F [r * SD + c] = v;
    FT[c * SD + r] = v;

    if (idx < SD) {
        if (idx < NDIM) {
            g[idx] = 1.0f;
        } else {
            float s = 0.0f;
            for (int j = 0; j < NDIM; ++j) s += LC[(idx - NDIM) * NDIM + j];
            g[idx] = s;
        }
    }
}

// ---------------------------------------------------------------------------
// h_k = F^k g for k=0..15; store Hmat[r][j] = h_{15-j}[r]  (256 x 16) so that
// d_b = Hmat @ w_block gives z_{16(b+1)} = F^16 z_{16b} + d_b.
// ---------------------------------------------------------------------------
__global__ void hChain(const float* __restrict__ FT, const float* __restrict__ g,
                       float* __restrict__ Hmat)
{
    __shared__ float hs[SD];
    const int r = threadIdx.x;
    hs[r] = g[r];
    __syncthreads();
    for (int k = 0; k < BS; ++k) {
        Hmat[r * BS + (BS - 1 - k)] = hs[r];
        if (k == BS - 1) break;
        float s = 0.0f;
        for (int j = 0; j < SD; ++j) s += FT[j * SD + r] * hs[j];  // (F hs)[r]
        __syncthreads();
        hs[r] = s;
        __syncthreads();
    }
}

// W[j][b] = w[16*b + j + 1]  (zero-padded past T-1), 16 x 256
__global__ void buildW(const float* __restrict__ w, float* __restrict__ W)
{
    const int idx = blockIdx.x * blockDim.x + threadIdx.x;    // 0 .. 4095
    const int j = idx >> 8, b = idx & 255;
    const int t = BS * b + j + 1;
    W[j * NB + b] = (t < TLEN) ? w[t] : 0.0f;
}

// ---------------------------------------------------------------------------
// Serial chain (the only sequential part left): 256 iterations of
//   z <- F16 z + d_b,  F16 register-resident (64 VGPRs x 1024 threads).
// Stores z at every block start into Zb (z_0 .. z_255).
// ---------------------------------------------------------------------------
__global__ void __launch_bounds__(1024)
seqChain(const float* __restrict__ F16, const float* __restrict__ D,
         float* __restrict__ Zb)
{
    const int tid = threadIdx.x;
    const int r = tid & 255;        // output row
    const int q = tid >> 8;         // quarter of the inner dim
    float freg[64];
#pragma unroll
    for (int j = 0; j < 64; ++j) freg[j] = F16[r * SD + q * 64 + j];

    __shared__ float zs[SD];
    __shared__ float part[1024];
    if (tid < SD) zs[tid] = 0.0f;
    __syncthreads();

    for (int b = 0; b < NB; ++b) {
        if (q == 0) Zb[b * SD + r] = zs[r];
        float s = 0.0f;
#pragma unroll
        for (int j = 0; j < 64; ++j) s += freg[j] * zs[q * 64 + j];
        part[tid] = s;
        __syncthreads();
        if (q == 0)
            zs[r] = part[r] + part[256 + r] + part[512 + r] + part[768 + r]
                  + D[r * NB + b];
        __syncthreads();
    }
}

// ---------------------------------------------------------------------------
// Per-block replay + cost (fully parallel across 256 blocks):
//   z_{16b+i} = F z_{16b+i-1} + w_t g ;  cost_t = x'Qx + xh'(K'RK)xh.
// Q and K'RK are symmetric -> column-indexed reads are coalesced.
// ---------------------------------------------------------------------------
__global__ void costKernel(const float* __restrict__ FT, const float* __restrict__ g,
                           const float* __restrict__ Q,  const float* __restrict__ KRK,
                           const float* __restrict__ Zb, const float* __restrict__ w,
                           float* __restrict__ out)
{
    const int b = blockIdx.x, r = threadIdx.x;
    __shared__ float xs[SD];
    __shared__ float red[SD];
    xs[r] = Zb[b * SD + r];
    __syncthreads();
    const float gr = g[r];

    for (int i = 1; i <= BS; ++i) {
        const int t = BS * b + i;
        const float wt = (t < TLEN) ? w[t] : 0.0f;
        float s = 0.0f;
#pragma unroll 8
        for (int j = 0; j < SD; ++j) s += FT[j * SD + r] * xs[j];   // (F x)[r]
        s += wt * gr;
        __syncthreads();
        xs[r] = s;
        __syncthreads();

        float qv = 0.0f;
        if (r < NDIM) {                       // x' Q x   (Q symmetric)
#pragma unroll 8
            for (int j = 0; j < NDIM; ++j) qv += Q[j * NDIM + r] * xs[j];
        } else {                              // xh' K'RK xh (symmetric)
            const int rr = r - NDIM;
#pragma unroll 8
            for (int j = 0; j < NDIM; ++j) qv += KRK[j * NDIM + rr] * xs[NDIM + j];
        }
        red[r] = qv * xs[r];
        __syncthreads();
        for (int off = NDIM; off > 0; off >>= 1) {
            if (r < off) red[r] += red[r + off];
            __syncthreads();
        }
        if (r == 0 && t < TLEN) out[t] = red[0];
        __syncthreads();
    }
    if (b == 0 && r == 0) out[0] = 0.0f;
}

// ---------------------------------------------------------------------------
struct GemmArgs {
    const float* A; int lda; bool tA;
    const float* B; int ldb;
    float* C; int ldc;
    const float* Ca; int ldca; float as;
    int M, N, K; float alpha;
};

static void launch_gemm(hipStream_t stream, const GemmArgs& a)
{
    const int tiles = (a.M / 16) * (a.N / 16);
    const int blocks = (tiles + 3) / 4;
    dim3 grd(blocks), blk(128);
#define GEMM_CALL(KD, TAv, ADDCv)                                               \
    wgemm_t<KD, TAv, ADDCv><<<grd, blk, 0, stream>>>(                           \
        a.A, a.lda, a.B, a.ldb, a.C, a.ldc, a.Ca, a.ldca, a.as, a.M, a.N, a.alpha)
    if (a.tA) {                      // only used with K == 32, no addend
        GEMM_CALL(32, true, false);
    } else if (a.Ca) {               // only used with K == 128
        GEMM_CALL(128, false, true);
    } else {
        switch (a.K) {
        case 16:  GEMM_CALL(16,  false, false); break;
        case 32:  GEMM_CALL(32,  false, false); break;
        case 64:  GEMM_CALL(64,  false, false); break;
        case 128: GEMM_CALL(128, false, false); break;
        default:  GEMM_CALL(256, false, false); break;
        }
    }
#undef GEMM_CALL
}

extern "C" void kernel_launch(void* const* d_in, const int* in_sizes, int n_in,
                              void* d_out, int out_size, void* d_ws, size_t ws_size,
                              hipStream_t stream)
{
    const float* A  = (const float*)d_in[0];   // 128x128
    const float* B  = (const float*)d_in[1];   // 128x32
    const float* Cm = (const float*)d_in[2];   // 64x128
    const float* Q  = (const float*)d_in[3];   // 128x128 (symmetric)
    const float* R  = (const float*)d_in[4];   // 32x32   (symmetric)
    const float* K  = (const float*)d_in[5];   // 32x128
    const float* L  = (const float*)d_in[6];   // 128x64
    const float* w  = (const float*)d_in[7];   // 4096
    float* out = (float*)d_out;

    float* ws   = (float*)d_ws;
    float* LC   = ws;             // 128x128
    float* BK   = LC   + 16384;   // 128x128
    float* LCA  = BK   + 16384;   // 128x128
    float* IB   = LCA  + 16384;   // 128x32   (B + LC B)
    float* IBK  = IB   + 4096;    // 128x128
    float* RK   = IBK  + 16384;   // 32x128
    float* KRK  = RK   + 4096;    // 128x128
    float* F    = KRK  + 16384;   // 256x256
    float* FT   = F    + 65536;   // 256x256
    float* F2   = FT   + 65536;
    float* F4   = F2   + 65536;
    float* F8   = F4   + 65536;
    float* F16  = F8   + 65536;
    float* g    = F16  + 65536;   // 256
    float* Hmat = g    + 256;     // 256x16
    float* W    = Hmat + 4096;    // 16x256
    float* D    = W    + 4096;    // 256x256
    float* Zb   = D    + 65536;   // 256x256  (z at block starts)

    // ---- small fused-matrix products (all WMMA) ----
    launch_gemm(stream, {L,  PDIM, false, Cm, NDIM, LC,  NDIM, nullptr, 0, 0.f, NDIM, NDIM, PDIM, 1.f}); // LC = L@C
    launch_gemm(stream, {B,  MDIM, false, K,  NDIM, BK,  NDIM, nullptr, 0, 0.f, NDIM, NDIM, MDIM, 1.f}); // BK = B@K
    launch_gemm(stream, {LC, NDIM, false, A,  NDIM, LCA, NDIM, nullptr, 0, 0.f, NDIM, NDIM, NDIM, 1.f}); // LCA = LC@A
    launch_gemm(stream, {LC, NDIM, false, B,  MDIM, IB,  MDIM, B, MDIM, 1.f,    NDIM, MDIM, NDIM, 1.f}); // IB = LC@B + B
    launch_gemm(stream, {IB, MDIM, false, K,  NDIM, IBK, NDIM, nullptr, 0, 0.f, NDIM, NDIM, MDIM, 1.f}); // IBK = IB@K
    launch_gemm(stream, {R,  MDIM, false, K,  NDIM, RK,  NDIM, nullptr, 0, 0.f, MDIM, NDIM, MDIM, 1.f}); // RK = R@K
    launch_gemm(stream, {K,  NDIM, true,  RK, NDIM, KRK, NDIM, nullptr, 0, 0.f, NDIM, NDIM, MDIM, 1.f}); // KRK = K'@RK

    assembleF<<<256, 256, 0, stream>>>(A, LC, BK, LCA, IBK, F, FT, g);

    // ---- F^16 by repeated squaring (WMMA 256^3 GEMMs) ----
    launch_gemm(stream, {F,  SD, false, F,  SD, F2,  SD, nullptr, 0, 0.f, SD, SD, SD, 1.f});
    launch_gemm(stream, {F2, SD, false, F2, SD, F4,  SD, nullptr, 0, 0.f, SD, SD, SD, 1.f});
    launch_gemm(stream, {F4, SD, false, F4, SD, F8,  SD, nullptr, 0, 0.f, SD, SD, SD, 1.f});
    launch_gemm(stream, {F8, SD, false, F8, SD, F16, SD, nullptr, 0, 0.f, SD, SD, SD, 1.f});

    // ---- per-block drive vectors for the whole trajectory in one GEMM ----
    hChain<<<1, 256, 0, stream>>>(FT, g, Hmat);
    buildW<<<16, 256, 0, stream>>>(w, W);
    launch_gemm(stream, {Hmat, BS, false, W, NB, D, NB, nullptr, 0, 0.f, SD, NB, BS, 1.f});  // D = H@W

    // ---- 256-step serial chain, then fully parallel cost replay ----
    seqChain<<<1, 1024, 0, stream>>>(F16, D, Zb);
    costKernel<<<NB, 256, 0, stream>>>(FT, g, Q, KRK, Zb, w, out);
}